// YOLOv3LossV2_71133248356573
// MI455X (gfx1250) — compile-verified
//
#include <hip/hip_runtime.h>
#include <math.h>

#define NCLS 80
#define NL 50
#define NBATCH 16
#define LOG2C 0.69314718055994531f

// Per-label precomputed record (48 bytes, 12 dwords)
struct Rec {
    float tx, ty, tw, th;   // truth box in grid units
    int   lin;              // matched cell linear index (a*H + j)*W + i, or -1
    int   cls;              // class id
    float tcx, tcy;         // xy targets (fractional cell offset)
    float lw, lh;           // wh targets (log-space)
    float sval;             // sqrt(2 - tw*th/(H*W))
    float pad;
};
#define REC_DW (NL * (int)(sizeof(Rec) / 4))   // 600 dwords

__constant__ float c_anchors[18] = {
    10.f,13.f, 16.f,30.f, 33.f,23.f, 30.f,61.f, 62.f,45.f,
    59.f,119.f, 116.f,90.f, 156.f,198.f, 373.f,326.f};

typedef __attribute__((ext_vector_type(2))) float v2f;
typedef __attribute__((ext_vector_type(8))) float v8f;
typedef __attribute__((ext_vector_type(4))) unsigned int u32x4;
typedef __attribute__((ext_vector_type(8))) int i32x8;
typedef __attribute__((ext_vector_type(4))) int i32x4;

__device__ __forceinline__ float bce(float x, float z) {
    // max(x,0) - x*z + log1p(exp(-|x|))
    return fmaxf(x, 0.f) - x * z + log1pf(expf(-fabsf(x)));
}

__global__ void zero_kernel(float* o) {
    if (threadIdx.x == 0 && blockIdx.x == 0) o[0] = 0.f;
}

// One thread per (layer, batch, label): 3*16*50 = 2400
__global__ void prep_kernel(const float* __restrict__ targets, Rec* __restrict__ recs) {
    int idx = blockIdx.x * blockDim.x + threadIdx.x;
    if (idx >= 3 * NBATCH * NL) return;
    int layer = idx / (NBATCH * NL);
    int r = idx - layer * (NBATCH * NL);
    int b = r / NL, l = r - (r / NL) * NL;

    int W = (layer == 0) ? 19 : (layer == 1) ? 38 : 76;
    float stride = (float)(32 >> layer);
    int group = 2 - layer;  // anch_mask[0] // 3

    const float* lab = targets + (size_t)(b * NL + l) * 5;
    float Wf = (float)W;
    float tx = lab[1] * Wf, ty = lab[2] * Wf;
    float tw = lab[3] * Wf, th = lab[4] * Wf;
    float areaT = tw * th;

    // argmax anchor IoU over 9 reference anchors (first max wins, like jnp.argmax)
    int best = 0; float bestIoU = -1.f;
    for (int k = 0; k < 9; ++k) {
        float rw = c_anchors[2 * k] / stride, rh = c_anchors[2 * k + 1] / stride;
        float mnx = fminf(tw, rw), mny = fminf(th, rh);
        float inter = mnx * mny;  // all positive -> en == 1
        float iou = inter / (areaT + rw * rh - inter);
        if (iou > bestIoU) { bestIoU = iou; best = k; }
    }
    int best_n = best % 3;
    bool matched = (best / 3) == group;
    int ti = (int)tx, tj = (int)ty;

    Rec rec;
    rec.tx = tx; rec.ty = ty; rec.tw = tw; rec.th = th;
    rec.lin = matched ? (best_n * W + tj) * W + ti : -1;
    rec.cls = (int)lab[0];
    rec.tcx = tx - (float)ti;
    rec.tcy = ty - (float)tj;
    float max_ = c_anchors[2 * (group * 3 + best_n)] / stride;      // m_anch[best_n].x
    float may_ = c_anchors[2 * (group * 3 + best_n) + 1] / stride;  // m_anch[best_n].y
    rec.lw = logf(tw / max_ + 1e-16f);
    rec.lh = logf(th / may_ + 1e-16f);
    rec.sval = sqrtf(2.0f - areaT / (Wf * Wf));
    rec.pad = 0.f;
    recs[layer * (NBATCH * NL) + b * NL + l] = rec;
}

// Block = 256 threads (8 wave32). Grid: x = cell tiles, y = batch.
__global__ __launch_bounds__(256) void loss_kernel(
    const float* __restrict__ out, const Rec* __restrict__ recs,
    float* __restrict__ loss, int W,
    float m0x, float m0y, float m1x, float m1y, float m2x, float m2y) {
    __shared__ Rec sh[NL];
    __shared__ float red[16];

    const int tid = threadIdx.x;
    const int b = blockIdx.y;
    const int HW = W * W;
    const int total = 3 * HW;

    // Stage the 50 truth records into LDS with the Tensor Data Mover:
    // one wave-issued DMA (600-dword 1-D tile), waited on TENSORcnt.
    if (tid == 0) {
        __builtin_prefetch(recs + b * NL, 0, 0);  // global_prefetch_b8
        unsigned long long gaddr = (unsigned long long)(uintptr_t)(recs + b * NL);
        unsigned int laddr = (unsigned int)(uintptr_t)(void*)sh;  // LDS aperture: low 32 bits = LDS offset
        u32x4 g0;
        g0.x = 1u;                                  // count=1 valid descriptor, user mode
        g0.y = laddr;                               // lds_addr (bytes)
        g0.z = (unsigned int)gaddr;                 // global_addr[31:0]
        g0.w = (unsigned int)((gaddr >> 32) & 0x1FFFFFFu) | (2u << 30);  // addr[56:32] | type=2
        i32x8 g1;
        g1[0] = (2 << 16);                          // workgroup_mask=0, data_size=4B
        g1[1] = (REC_DW & 0xFFFF) << 16;            // tensor_dim0[15:0]
        g1[2] = (REC_DW >> 16) | (1 << 16);         // tensor_dim0[31:16] | tensor_dim1[15:0]=1
        g1[3] = (REC_DW & 0xFFFF) << 16;            // tensor_dim1[31:16]=0 | tile_dim0=600
        g1[4] = 1;                                  // tile_dim1=1, tile_dim2=0
        g1[5] = REC_DW;                             // tensor_dim0_stride[31:0]
        g1[6] = 0;                                  // stride hi | tensor_dim1_stride lo
        g1[7] = 0;
        i32x4 gz = {0, 0, 0, 0};                    // groups 2/3 unused (<=2-D tensor)
        i32x8 gz8 = {0, 0, 0, 0, 0, 0, 0, 0};
        __builtin_amdgcn_tensor_load_to_lds(g0, g1, gz, gz, gz8, 0);
        __builtin_amdgcn_s_wait_tensorcnt(0);       // s_wait_tensorcnt 0
    }
    __syncthreads();

    int cell = blockIdx.x * 256 + tid;
    float acc = 0.f;
    if (cell < total) {
        int a = cell / HW;
        int rem = cell - a * HW;
        int h = rem / W, w = rem - h * W;
        const float* base = out + (size_t)(b * 255 + a * 85) * HW + rem;
        // Only channels 0..4 are needed for every cell.
        float o0 = base[0];
        float o1 = base[(size_t)HW];
        float o2 = base[(size_t)2 * HW];
        float o3 = base[(size_t)3 * HW];
        float o4 = base[(size_t)4 * HW];

        float ax = (a == 0) ? m0x : (a == 1) ? m1x : m2x;
        float ay = (a == 0) ? m0y : (a == 1) ? m1y : m2y;
        float px = 1.f / (1.f + expf(-o0)) + (float)w;
        float py = 1.f / (1.f + expf(-o1)) + (float)h;
        float pw = expf(o2) * ax;
        float ph = expf(o3) * ay;
        float hx = 0.5f * pw, hy = 0.5f * ph;
        float areaP = pw * ph;

        float maxiou = -1.f;
        int found = -1;  // last matching label wins (scatter-set semantics)
        for (int t = 0; t < NL; ++t) {
            float txx = sh[t].tx, tyy = sh[t].ty, tww = sh[t].tw, thh = sh[t].th;
            float tlx = fmaxf(px - hx, txx - 0.5f * tww);
            float tly = fmaxf(py - hy, tyy - 0.5f * thh);
            float brx = fminf(px + hx, txx + 0.5f * tww);
            float bry = fminf(py + hy, tyy + 0.5f * thh);
            float en = ((tlx < brx) && (tly < bry)) ? 1.f : 0.f;
            float inter = (brx - tlx) * (bry - tly) * en;
            float iou = inter / (areaP + tww * thh - inter);
            maxiou = fmaxf(maxiou, iou);
            if (sh[t].lin == cell) found = t;
        }
        bool fnd = found >= 0;
        float objv = fnd ? 1.f : ((maxiou <= 0.7f) ? 1.f : 0.f);

        // objectness loss: ignored cells contribute bce(0,0)=log2
        acc += (objv == 0.f) ? LOG2C : bce(o4, fnd ? 1.f : 0.f);

        if (fnd) {
            Rec rc = sh[found];
            float sc2 = rc.sval * rc.sval;
            acc += (bce(o0, rc.tcx) + bce(o1, rc.tcy)) * sc2;
            float dw = o2 - rc.lw, dh = o3 - rc.lh;
            acc += 0.5f * sc2 * (dw * dw + dh * dh);
            const float* cbase = base + (size_t)5 * HW;
            for (int c = 0; c < NCLS; ++c) {
                float oc = cbase[(size_t)c * HW];
                acc += bce(oc, (c == rc.cls) ? 1.f : 0.f);
            }
        } else {
            // unselected: class loss is exactly 80 * log(2); xy/wh are 0
            acc += (float)NCLS * LOG2C;
        }
    }

    // Per-wave reduction via matrix unit: D = A x ones.
    // A(16x4): lane m<16 -> A[m][0]=acc, lane m>=16 -> A[m-16][2]=acc.
    // D[m][n] = acc_m + acc_{m+16}; lane 0 holds rows 0..7, lane 16 rows 8..15.
    v2f av; av.x = acc; av.y = 0.f;
    v2f bv; bv.x = 1.f; bv.y = 1.f;
    v8f cv = {};
    v8f dv = __builtin_amdgcn_wmma_f32_16x16x4_f32(
        false, av, false, bv, (short)0, cv, false, false);
    float s = 0.f;
#pragma unroll
    for (int i = 0; i < 8; ++i) s += dv[i];

    int lane = tid & 31, wave = tid >> 5;
    if ((lane & 15) == 0) red[wave * 2 + (lane >> 4)] = s;
    __syncthreads();
    if (tid == 0) {
        float tot = 0.f;
#pragma unroll
        for (int i = 0; i < 16; ++i) tot += red[i];
        atomicAdd(loss, tot);
    }
}

extern "C" void kernel_launch(void* const* d_in, const int* in_sizes, int n_in,
                              void* d_out, int out_size, void* d_ws, size_t ws_size,
                              hipStream_t stream) {
    const float* outs[3] = {(const float*)d_in[0], (const float*)d_in[1], (const float*)d_in[2]};
    const float* targets = (const float*)d_in[3];
    float* loss = (float*)d_out;
    Rec* recs = (Rec*)d_ws;

    zero_kernel<<<1, 32, 0, stream>>>(loss);
    prep_kernel<<<(3 * NBATCH * NL + 255) / 256, 256, 0, stream>>>(targets, recs);

    static const int   Ws[3] = {19, 38, 76};
    static const float strides[3] = {32.f, 16.f, 8.f};
    static const float A[18] = {10, 13, 16, 30, 33, 23, 30, 61, 62, 45,
                                59, 119, 116, 90, 156, 198, 373, 326};
    for (int l = 0; l < 3; ++l) {
        int W = Ws[l];
        int group = 2 - l;
        float m[6];
        for (int n = 0; n < 3; ++n) {
            m[2 * n]     = A[2 * (group * 3 + n)]     / strides[l];
            m[2 * n + 1] = A[2 * (group * 3 + n) + 1] / strides[l];
        }
        int total = 3 * W * W;
        dim3 grid((total + 255) / 256, NBATCH);
        loss_kernel<<<grid, 256, 0, stream>>>(outs[l], recs + l * NBATCH * NL, loss,
                                              W, m[0], m[1], m[2], m[3], m[4], m[5]);
    }
}